// GCNLayer_61392262529024
// MI455X (gfx1250) — compile-verified
//
#include <hip/hip_runtime.h>

// ---------------------------------------------------------------------------
// GNN 2-branch forward for MI455X (gfx1250, wave32).
//  - Edge aggregation: L2-resident scatter-add with hardware f32 atomics.
//  - Dense layers:     V_WMMA_F32_16X16X4_F32 (full f32 precision).
// ---------------------------------------------------------------------------

typedef __attribute__((ext_vector_type(2))) float v2f;
typedef __attribute__((ext_vector_type(8))) float v8f;

#define FX 128
#define FE 8

// ---------------------------------------------------------------------------
// Scatter-add of gathered node features: agg[dst] += x[src], one wave per edge.
// 32 lanes * float4 = 128 features. x table (25.6MB) is L2-resident -> gathers
// hit L2; atomics are resolved at L2.
// ---------------------------------------------------------------------------
__global__ __launch_bounds__(256) void scatter_x_kernel(
    const float* __restrict__ x, const int* __restrict__ ei,
    float* __restrict__ agg, int n_edges)
{
    const int wave = threadIdx.x >> 5;
    const int lane = threadIdx.x & 31;
    const int e = blockIdx.x * 8 + wave;
    if (e >= n_edges) return;
    const int src = ei[e];
    const int dst = ei[n_edges + e];
    const float4 v = ((const float4*)(x + (size_t)src * FX))[lane];
    float* p = agg + (size_t)dst * FX + lane * 4;
    unsafeAtomicAdd(p + 0, v.x);
    unsafeAtomicAdd(p + 1, v.y);
    unsafeAtomicAdd(p + 2, v.z);
    unsafeAtomicAdd(p + 3, v.w);
}

// ---------------------------------------------------------------------------
// Scatter-add of edge attributes: aggE[dst] += edge_attr[e]. One thread per
// (edge, feature). Layer-invariant: run once per branch.
// ---------------------------------------------------------------------------
__global__ __launch_bounds__(256) void scatter_e_kernel(
    const float* __restrict__ ea, const int* __restrict__ ei,
    float* __restrict__ aggE, int n_edges)
{
    const int t = blockIdx.x * blockDim.x + threadIdx.x;
    if (t >= n_edges * FE) return;
    const int e = t >> 3;
    const int f = t & 7;
    const int dst = ei[n_edges + e];
    unsafeAtomicAdd(aggE + (size_t)dst * FE + f, ea[(size_t)e * FE + f]);
}

// ---------------------------------------------------------------------------
// Y[row, col] = act( [A0 | A1][row, :] @ W[col, :]^T + bias[col] )
//   A0: [n_rows, 128], A1: [n_rows, k1] (k1 = 8 or 0), W: [128, 128+k1] row
//   major (nn.Linear layout), Y: [n_rows, 128].
// One wave = one 16x16 output tile via V_WMMA_F32_16X16X4_F32.
// 8 waves per block cover all 128 output columns of one 16-row tile.
//
// f32 WMMA VGPR layouts (ISA 7.12.2):
//   A 16x4 : lanes 0-15 -> M=lane,    v0=K0, v1=K1
//            lanes16-31 -> M=lane-16, v0=K2, v1=K3
//   B 4x16 : v0 = rows K0 (lanes 0-15) / K2 (lanes 16-31), v1 = K1 / K3
//   C 16x16: VGPR r = rows M=r (lanes 0-15) and M=r+8 (lanes 16-31)
// ---------------------------------------------------------------------------
__global__ __launch_bounds__(256) void gemm_wmma_kernel(
    const float* __restrict__ A0, const float* __restrict__ A1,
    const float* __restrict__ W, const float* __restrict__ bias,
    float* __restrict__ Y, int n_rows, int k1, int leaky)
{
    const int lane = threadIdx.x & 31;
    const int wtile = threadIdx.x >> 5;          // col tile 0..7
    const int half = lane >> 4;                  // 0 or 1
    const int l15 = lane & 15;
    const int row0 = blockIdx.x * 16;
    const int col0 = wtile * 16;
    const int K = FX + k1;                       // W row stride (136 or 128)

    int arow = row0 + l15;
    if (arow >= n_rows) arow = n_rows - 1;       // clamp (EXEC must stay all-1)

    const float* __restrict__ a0p = A0 + (size_t)arow * FX;
    const float* __restrict__ wp = W + (size_t)(col0 + l15) * K;

    v8f acc = {};
    // main K loop over the 128 node-feature columns
    for (int k = 0; k < FX; k += 4) {
        v2f a, b;
        const int ka = k + half * 2;
        a.x = a0p[ka];
        a.y = a0p[ka + 1];
        b.x = wp[ka];
        b.y = wp[ka + 1];
        acc = __builtin_amdgcn_wmma_f32_16x16x4_f32(
            false, a, false, b, (short)0, acc, false, false);
    }
    // tail: edge-feature columns (K = 128..128+k1)
    if (k1 > 0) {
        const float* __restrict__ a1p = A1 + (size_t)arow * FE;
        for (int k = FX; k < FX + k1; k += 4) {
            v2f a, b;
            const int ka = k + half * 2;
            a.x = a1p[ka - FX];
            a.y = a1p[ka - FX + 1];
            b.x = wp[ka];
            b.y = wp[ka + 1];
            acc = __builtin_amdgcn_wmma_f32_16x16x4_f32(
                false, a, false, b, (short)0, acc, false, false);
        }
    }

    const int col = col0 + l15;
    const float bv = bias[col];
#pragma unroll
    for (int r = 0; r < 8; ++r) {
        const int row = row0 + r + half * 8;
        if (row >= n_rows) continue;
        float v = acc[r] + bv;
        if (leaky) v = (v > 0.0f) ? v : 0.01f * v;
        Y[(size_t)row * FX + col] = v;
    }
}

// ---------------------------------------------------------------------------
// Host-side orchestration (all on `stream`; graph-capture safe).
// ---------------------------------------------------------------------------
extern "C" void kernel_launch(void* const* d_in, const int* in_sizes, int n_in,
                              void* d_out, int out_size, void* d_ws, size_t ws_size,
                              hipStream_t stream)
{
    const int n_nodes = in_sizes[0] / FX;     // 50000
    const int n_edges = in_sizes[1] / 2;      // 1600000
    const int L       = in_sizes[7] / FX;     // 3

    // workspace layout (floats): xA | xB | aggX | aggE
    float* ws   = (float*)d_ws;
    float* xA   = ws;
    float* xB   = xA + (size_t)n_nodes * FX;
    float* aggX = xB + (size_t)n_nodes * FX;
    float* aggE = aggX + (size_t)n_nodes * FX;

    const float* fcW = (const float*)d_in[10];
    const float* fcb = (const float*)d_in[11];

    const int row_tiles = (n_nodes + 15) / 16;

    for (int br = 0; br < 2; ++br) {
        const float* x0 = (const float*)d_in[br ? 3 : 0];
        const int*   ei = (const int*)  d_in[br ? 4 : 1];
        const float* ea = (const float*)d_in[br ? 5 : 2];
        const float* Wl = (const float*)d_in[br ? 8 : 6];  // [L,128,136]
        const float* bl = (const float*)d_in[br ? 9 : 7];  // [L,128]
        float* out = (float*)d_out + (size_t)br * n_nodes * FX;

        // layer-invariant edge-attr aggregation
        hipMemsetAsync(aggE, 0, (size_t)n_nodes * FE * sizeof(float), stream);
        {
            const int total = n_edges * FE;
            scatter_e_kernel<<<(total + 255) / 256, 256, 0, stream>>>(
                ea, ei, aggE, n_edges);
        }

        const float* x_cur = x0;
        for (int l = 0; l < L; ++l) {
            hipMemsetAsync(aggX, 0, (size_t)n_nodes * FX * sizeof(float), stream);
            scatter_x_kernel<<<(n_edges + 7) / 8, 256, 0, stream>>>(
                x_cur, ei, aggX, n_edges);
            float* x_next = (l & 1) ? xB : xA;
            gemm_wmma_kernel<<<row_tiles, 256, 0, stream>>>(
                aggX, aggE,
                Wl + (size_t)l * FX * (FX + FE), bl + (size_t)l * FX,
                x_next, n_nodes, FE, /*leaky=*/1);
            x_cur = x_next;
        }
        // final shared FC (K=128, no activation) straight into d_out
        gemm_wmma_kernel<<<row_tiles, 256, 0, stream>>>(
            x_cur, nullptr, fcW, fcb, out, n_nodes, /*k1=*/0, /*leaky=*/0);
    }
}